// OptimalModel_87213605913340
// MI455X (gfx1250) — compile-verified
//
#include <hip/hip_runtime.h>
#include <math.h>

typedef float v2f __attribute__((ext_vector_type(2)));
typedef float v8f __attribute__((ext_vector_type(8)));

#define B_ 16
#define N_ 1024
#define D_ 3072
#define KC 256
#define GAMMA_MIN_ -15.0f
#define GAMMA_MAX_ 10.0f

struct Scal { float inv_s2, sqrt1pv, alpha, sigma; };

__device__ inline Scal get_scalars(const float* __restrict__ ng_ptr) {
  float ng = ng_ptr[0];
  float gamma = ng * (GAMMA_MAX_ - GAMMA_MIN_) + GAMMA_MIN_;
  float eg = expf(gamma);           // e^gamma = 1/var_tilde = 1/scale^2
  float var_tilde = expf(-gamma);   // scale^2
  Scal s;
  s.inv_s2 = eg;
  s.sqrt1pv = sqrtf(1.0f + var_tilde);
  float variance = 1.0f / (1.0f + eg);   // sigmoid(-gamma)
  s.alpha = sqrtf(1.0f - variance);
  s.sigma = sqrtf(variance);
  return s;
}

// ---------------------------------------------------------------------------
// Kernel 1: Z[b,n] = inv_s2 * G[b,n] - 0.5*inv_s2*rowsum_t[n]
//   G = (mask * sqrt(1+vt) * in) . T^T   via V_WMMA_F32_16X16X4_F32
// Grid: 64 blocks (one 16-wide n tile each), 128 threads (4 waves split K).
// ---------------------------------------------------------------------------
__global__ __launch_bounds__(128) void k_logits(
    const float* __restrict__ in_t, const float* __restrict__ ng,
    const float* __restrict__ mask, const float* __restrict__ train,
    float* __restrict__ Z) {
  __shared__ float lds_t[KC][17];      // T chunk transposed [k][n], padded
  __shared__ float lds_s[B_][KC + 1];  // masked/scaled state chunk [b][k]
  __shared__ float lds_rows[16];       // rowsum_t for this n tile
  __shared__ float lds_red[4][32][8];  // cross-wave accumulator reduction

  const int tid  = threadIdx.x;
  const int wave = tid >> 5;
  const int lane = tid & 31;
  const int half = lane >> 4;
  const int l    = lane & 15;
  const int n0   = blockIdx.x * 16;

  const Scal sc = get_scalars(ng);

  if (tid < 16) lds_rows[tid] = 0.0f;

  // staging decomposition: 8 threads per row, 16 rows
  const int row = tid >> 3;   // 0..15 : n-within-tile for T, b for S
  const int kl  = tid & 7;    // 0..7

  float rs_acc = 0.0f;                 // partial m*t^2 for row n0+row
  v8f c = {0.f,0.f,0.f,0.f,0.f,0.f,0.f,0.f};

  for (int kb = 0; kb < D_; kb += KC) {
    __syncthreads();                   // LDS reuse from previous chunk
    const float* trow = train + (size_t)(n0 + row) * D_ + kb;
    const float* irow = in_t  + (size_t)row * D_ + kb;
#pragma unroll 4
    for (int i = 0; i < KC / 8; ++i) {
      int k = kl + i * 8;
      float mk = mask[kb + k];
      float tv = trow[k];
      lds_t[k][row] = tv;
      rs_acc += mk * tv * tv;
      lds_s[row][k] = irow[k] * mk * sc.sqrt1pv;
    }
    __syncthreads();
    // each wave covers its quarter of the chunk, 4 K per WMMA
    const int kw = wave * (KC / 4);
#pragma unroll 4
    for (int kk = 0; kk < KC / 4; kk += 4) {
      const int k4 = kw + kk;
      v2f a, b;
      // A 16x4 f32 frag: lanes 0-15 -> K=0,1 ; lanes 16-31 -> K=2,3
      a.x = lds_s[l][k4 + 2 * half];
      a.y = lds_s[l][k4 + 2 * half + 1];
      // B 4x16 f32 frag: VGPR0 rows {0,2}, VGPR1 rows {1,3} (half-split)
      b.x = lds_t[k4 + 2 * half][l];
      b.y = lds_t[k4 + 2 * half + 1][l];
      c = __builtin_amdgcn_wmma_f32_16x16x4_f32(
          /*neg_a=*/false, a, /*neg_b=*/false, b,
          /*c_mod=*/(short)0, c, /*reuse_a=*/false, /*reuse_b=*/false);
    }
  }

  atomicAdd(&lds_rows[row], rs_acc);   // ds_add_f32, 8 adders per n

#pragma unroll
  for (int r = 0; r < 8; ++r) lds_red[wave][lane][r] = c[r];
  __syncthreads();

  if (wave == 0) {
#pragma unroll
    for (int r = 0; r < 8; ++r) {
      float g = lds_red[0][lane][r] + lds_red[1][lane][r] +
                lds_red[2][lane][r] + lds_red[3][lane][r];
      // C/D layout: VGPR r -> M = r + 8*half, N = lane&15
      int b_idx = r + 8 * half;
      float z = sc.inv_s2 * g - 0.5f * sc.inv_s2 * lds_rows[l];
      Z[(size_t)b_idx * N_ + n0 + l] = z;
    }
  }
}

// ---------------------------------------------------------------------------
// Kernel 2: P[b,:] = softmax(Z[b,:]) over N=1024. Grid: 16 blocks x 256.
// ---------------------------------------------------------------------------
__global__ __launch_bounds__(256) void k_softmax(
    const float* __restrict__ Z, float* __restrict__ P) {
  __shared__ float red[256];
  const int b = blockIdx.x, tid = threadIdx.x;
  const float* z = Z + (size_t)b * N_;

  float m = -INFINITY;
  for (int j = tid; j < N_; j += 256) m = fmaxf(m, z[j]);
  red[tid] = m; __syncthreads();
  for (int s = 128; s > 0; s >>= 1) {
    if (tid < s) red[tid] = fmaxf(red[tid], red[tid + s]);
    __syncthreads();
  }
  m = red[0]; __syncthreads();

  float sum = 0.0f;
  for (int j = tid; j < N_; j += 256) sum += expf(z[j] - m);
  red[tid] = sum; __syncthreads();
  for (int s = 128; s > 0; s >>= 1) {
    if (tid < s) red[tid] += red[tid + s];
    __syncthreads();
  }
  float inv = 1.0f / red[0];
  for (int j = tid; j < N_; j += 256)
    P[(size_t)b * N_ + j] = expf(z[j] - m) * inv;
}

// ---------------------------------------------------------------------------
// Kernel 3: denoised = P . T (16x3072, K=1024) via WMMA; fused epilogue
//   out = mask*(in - alpha*mask*dn)/sigma
// Grid: 48 blocks x 128 threads; each wave owns one 16-wide d tile.
// ---------------------------------------------------------------------------
__global__ __launch_bounds__(128) void k_denoise(
    const float* __restrict__ in_t, const float* __restrict__ ng,
    const float* __restrict__ mask, const float* __restrict__ train,
    const float* __restrict__ P, float* __restrict__ out) {
  const int tid  = threadIdx.x;
  const int wave = tid >> 5;
  const int lane = tid & 31;
  const int half = lane >> 4;
  const int l    = lane & 15;
  const int d0   = (blockIdx.x * 4 + wave) * 16;   // 192 tiles total

  const Scal sc = get_scalars(ng);
  v8f c = {0.f,0.f,0.f,0.f,0.f,0.f,0.f,0.f};

  for (int k = 0; k < N_; k += 4) {
    v2f a, b;
    const float* pr = P + (size_t)l * N_ + k + 2 * half;
    a.x = pr[0];                                        // A = prob (16 x K)
    a.y = pr[1];
    b.x = train[(size_t)(k + 2 * half)     * D_ + d0 + l];  // B = T rows
    b.y = train[(size_t)(k + 2 * half + 1) * D_ + d0 + l];
    c = __builtin_amdgcn_wmma_f32_16x16x4_f32(
        false, a, false, b, (short)0, c, false, false);
  }

  const float inv_sig = 1.0f / sc.sigma;
#pragma unroll
  for (int r = 0; r < 8; ++r) {
    int b_idx = r + 8 * half;
    int d = d0 + l;
    float mval = mask[d];
    float dn = c[r] * mval;            // denoised = (sum) * mask
    float eps = (in_t[(size_t)b_idx * D_ + d] - sc.alpha * dn) * inv_sig;
    out[(size_t)b_idx * D_ + d] = eps * mval;
  }
}

extern "C" void kernel_launch(void* const* d_in, const int* in_sizes, int n_in,
                              void* d_out, int out_size, void* d_ws, size_t ws_size,
                              hipStream_t stream) {
  const float* in_t  = (const float*)d_in[0];   // [16,3,32,32]
  const float* ng    = (const float*)d_in[1];   // [1]
  const float* mask  = (const float*)d_in[2];   // [1,3,32,32]
  const float* train = (const float*)d_in[3];   // [1024,3,32,32]
  float* out = (float*)d_out;                   // [16,3,32,32]

  float* Z = (float*)d_ws;          // 16*1024 floats
  float* P = Z + (size_t)B_ * N_;   // 16*1024 floats (128 KB total scratch)

  k_logits <<<N_ / 16, 128, 0, stream>>>(in_t, ng, mask, train, Z);
  k_softmax<<<B_,      256, 0, stream>>>(Z, P);
  k_denoise<<<48,      128, 0, stream>>>(in_t, ng, mask, train, P, out);
}